// GATConv_22565758173853
// MI455X (gfx1250) — compile-verified
//
#include <hip/hip_runtime.h>

#define F_DIM 256
#define HD 128
#define H_HEADS 4
#define NEG_SLOPE 0.2f

typedef __attribute__((ext_vector_type(2))) float v2f;
typedef __attribute__((ext_vector_type(8))) float v8f;

// Order-preserving float->uint encoding so atomicMax(uint) == float max.
__device__ __forceinline__ unsigned enc_f32(float f) {
    unsigned u = __float_as_uint(f);
    return (u & 0x80000000u) ? ~u : (u | 0x80000000u);
}
__device__ __forceinline__ float dec_f32(unsigned u) {
    unsigned b = (u & 0x80000000u) ? (u & 0x7FFFFFFFu) : ~u;
    return __uint_as_float(b);
}
__device__ __forceinline__ float lrelu(float x) {
    return x >= 0.0f ? x : NEG_SLOPE * x;
}

// ---------------------------------------------------------------------------
// Kernel 1: h = feat @ fc_w^T  via V_WMMA_F32_16X16X4_F32 (exact fp32).
// Block = 256 threads = 8 waves; wave w computes the 16x16 tile at columns
// [16w, 16w+16) for the 16 rows of this block. K loop over F=256 in steps of 4.
// A layout (16x4 f32): lane<16 -> K=0,1 in v[0],v[1]; lane>=16 -> K=2,3.
// B layout mirrors A; B[k][col] = fc_w[col*F + k] (fc_w is (HD, F) row-major).
// D layout: VGPR j: lanes 0-15 -> M=j, lanes 16-31 -> M=j+8; N = lane&15.
// ---------------------------------------------------------------------------
__global__ void gat_gemm_wmma(const float* __restrict__ feat,
                              const float* __restrict__ fcw,
                              float* __restrict__ h, int N) {
    const int lane = threadIdx.x & 31;
    const int wave = threadIdx.x >> 5;
    const int tileM = blockIdx.x * 16;
    const int r = lane & 15;
    int row = tileM + r;
    int rowc = row < N ? row : 0;            // clamp (no divergence before WMMA)
    const int col = wave * 16 + r;           // HD = 128 = 8 waves * 16
    const int ksel = (lane >> 4) << 1;       // 0 or 2

    const float* ap = feat + (size_t)rowc * F_DIM + ksel;
    const float* bp = fcw  + (size_t)col  * F_DIM + ksel;

    v8f acc = {};
#pragma unroll 8
    for (int k = 0; k < F_DIM; k += 4) {
        v2f a = *(const v2f*)(ap + k);
        v2f b = *(const v2f*)(bp + k);
        acc = __builtin_amdgcn_wmma_f32_16x16x4_f32(
            false, a, false, b, (short)0, acc, false, false);
    }

    const int mbase = tileM + ((lane >> 4) << 3);
    float* hp = h + (size_t)mbase * HD + col;
    if (tileM + 16 <= N) {                   // full tile: branch-free stores
#pragma unroll
        for (int j = 0; j < 8; ++j) hp[(size_t)j * HD] = acc[j];
    } else {                                 // tail tile (not hit for N%16==0)
#pragma unroll
        for (int j = 0; j < 8; ++j)
            if (mbase + j < N) hp[(size_t)j * HD] = acc[j];
    }
}

// ---------------------------------------------------------------------------
// Kernel 2: el[n,h] = <h[n,h,:], attn_l[h,:]>, er likewise. One wave per node,
// lane holds 4 channels (32 lanes * 4 = 128); heads = groups of 8 lanes.
// ---------------------------------------------------------------------------
__global__ void gat_logits(const float* __restrict__ h,
                           const float* __restrict__ attn_l,
                           const float* __restrict__ attn_r,
                           float* __restrict__ el, float* __restrict__ er, int N) {
    const int lane = threadIdx.x & 31;
    const int wave = threadIdx.x >> 5;
    const int n = blockIdx.x * (blockDim.x >> 5) + wave;
    if (n >= N) return;
    const float4 hh = *(const float4*)(h + (size_t)n * HD + lane * 4);
    const float4 al = *(const float4*)(attn_l + lane * 4);
    const float4 ar = *(const float4*)(attn_r + lane * 4);
    float pl = hh.x * al.x + hh.y * al.y + hh.z * al.z + hh.w * al.w;
    float pr = hh.x * ar.x + hh.y * ar.y + hh.z * ar.z + hh.w * ar.w;
#pragma unroll
    for (int m = 1; m < 8; m <<= 1) {        // reduce within the 8-lane head group
        pl += __shfl_xor(pl, m, 32);
        pr += __shfl_xor(pr, m, 32);
    }
    if ((lane & 7) == 0) {
        int head = lane >> 3;
        el[n * H_HEADS + head] = pl;
        er[n * H_HEADS + head] = pr;
    }
}

// ---------------------------------------------------------------------------
// Kernel 3: init out = bias (scatter then accumulates on top), m = enc(-inf~),
// denom = 0. d_out/d_ws are poisoned by the harness so this must run each call.
// ---------------------------------------------------------------------------
__global__ void gat_init(float* __restrict__ out, const float* __restrict__ bias,
                         unsigned* __restrict__ menc, float* __restrict__ denom,
                         int N) {
    size_t i = (size_t)blockIdx.x * blockDim.x + threadIdx.x;
    if (i < (size_t)N * HD) out[i] = bias[i & (HD - 1)];
    if (i < (size_t)N * H_HEADS) { menc[i] = 0u; denom[i] = 0.0f; }
}

// ---------------------------------------------------------------------------
// Kernel 4: segment max over edges by dst (atomicMax on monotone uint enc).
// el/er rows loaded as one b128 each instead of 4 scalar loads.
// ---------------------------------------------------------------------------
__global__ void gat_edge_max(const int* __restrict__ src, const int* __restrict__ dst,
                             const float* __restrict__ el, const float* __restrict__ er,
                             unsigned* __restrict__ menc, int E) {
    int e = blockIdx.x * blockDim.x + threadIdx.x;
    if (e >= E) return;
    int s = src[e], d = dst[e];
    const float4 es = *(const float4*)(el + (size_t)s * H_HEADS);
    const float4 ed = *(const float4*)(er + (size_t)d * H_HEADS);
    unsigned* mp = menc + (size_t)d * H_HEADS;
    atomicMax(mp + 0, enc_f32(lrelu(es.x + ed.x)));
    atomicMax(mp + 1, enc_f32(lrelu(es.y + ed.y)));
    atomicMax(mp + 2, enc_f32(lrelu(es.z + ed.z)));
    atomicMax(mp + 3, enc_f32(lrelu(es.w + ed.w)));
}

// ---------------------------------------------------------------------------
// Kernel 5: denom[d,h] += exp(e - m[d,h])  (recompute e; cheaper than storing
// an E x H buffer, and el/er/m all live in L2).
// ---------------------------------------------------------------------------
__global__ void gat_edge_denom(const int* __restrict__ src, const int* __restrict__ dst,
                               const float* __restrict__ el, const float* __restrict__ er,
                               const unsigned* __restrict__ menc,
                               float* __restrict__ denom, int E) {
    int e = blockIdx.x * blockDim.x + threadIdx.x;
    if (e >= E) return;
    int s = src[e], d = dst[e];
    const float4 es = *(const float4*)(el + (size_t)s * H_HEADS);
    const float4 ed = *(const float4*)(er + (size_t)d * H_HEADS);
    const uint4  mu = *(const uint4*)(menc + (size_t)d * H_HEADS);
    float* dp = denom + (size_t)d * H_HEADS;
    atomicAdd(dp + 0, __expf(lrelu(es.x + ed.x) - dec_f32(mu.x)));
    atomicAdd(dp + 1, __expf(lrelu(es.y + ed.y) - dec_f32(mu.y)));
    atomicAdd(dp + 2, __expf(lrelu(es.z + ed.z) - dec_f32(mu.z)));
    atomicAdd(dp + 3, __expf(lrelu(es.w + ed.w) - dec_f32(mu.w)));
}

// ---------------------------------------------------------------------------
// Kernel 6: out[dst] += alpha * h[src]. One wave per edge; lane carries 4
// channels (float4 gather from h, 4 f32 atomics into out). h and out are both
// L2-resident (25.6 MB each vs 192 MB L2), so this is an L2-atomic-rate kernel.
// ---------------------------------------------------------------------------
__global__ void gat_scatter(const int* __restrict__ src, const int* __restrict__ dst,
                            const float* __restrict__ el, const float* __restrict__ er,
                            const unsigned* __restrict__ menc,
                            const float* __restrict__ denom,
                            const float* __restrict__ h, float* __restrict__ out,
                            int E) {
    const int lane = threadIdx.x & 31;
    const int wave = threadIdx.x >> 5;
    const int e = blockIdx.x * (blockDim.x >> 5) + wave;
    if (e >= E) return;
    const int s = src[e], d = dst[e];
    const int head = lane >> 3;
    float v = lrelu(el[s * H_HEADS + head] + er[d * H_HEADS + head]);
    float m = dec_f32(menc[d * H_HEADS + head]);
    float alpha = __expf(v - m) / denom[d * H_HEADS + head];
    const float4 hh = *(const float4*)(h + (size_t)s * HD + lane * 4);
    float* o = out + (size_t)d * HD + lane * 4;
    atomicAdd(o + 0, hh.x * alpha);
    atomicAdd(o + 1, hh.y * alpha);
    atomicAdd(o + 2, hh.z * alpha);
    atomicAdd(o + 3, hh.w * alpha);
}

extern "C" void kernel_launch(void* const* d_in, const int* in_sizes, int n_in,
                              void* d_out, int out_size, void* d_ws, size_t ws_size,
                              hipStream_t stream) {
    const float* feat  = (const float*)d_in[0];
    const float* fcw   = (const float*)d_in[1];
    const float* attnl = (const float*)d_in[2];
    const float* attnr = (const float*)d_in[3];
    const float* bias  = (const float*)d_in[4];
    const int*   src   = (const int*)d_in[5];
    const int*   dst   = (const int*)d_in[6];
    float* out = (float*)d_out;

    const int N = in_sizes[0] / F_DIM;   // feat is (N, 256)
    const int E = in_sizes[5];

    // Workspace layout (floats): h | el | er | menc | denom  (~28.8 MB)
    float*    h     = (float*)d_ws;
    float*    el    = h  + (size_t)N * HD;
    float*    er    = el + (size_t)N * H_HEADS;
    unsigned* menc  = (unsigned*)(er + (size_t)N * H_HEADS);
    float*    denom = (float*)(menc + (size_t)N * H_HEADS);

    const int rowTiles = (N + 15) / 16;
    gat_gemm_wmma<<<rowTiles, 256, 0, stream>>>(feat, fcw, h, N);
    gat_logits<<<(N + 7) / 8, 256, 0, stream>>>(h, attnl, attnr, el, er, N);

    size_t initTotal = (size_t)N * HD;
    gat_init<<<(unsigned)((initTotal + 255) / 256), 256, 0, stream>>>(out, bias, menc, denom, N);

    gat_edge_max<<<(E + 255) / 256, 256, 0, stream>>>(src, dst, el, er, menc, E);
    gat_edge_denom<<<(E + 255) / 256, 256, 0, stream>>>(src, dst, el, er, menc, denom, E);
    gat_scatter<<<(E + 7) / 8, 256, 0, stream>>>(src, dst, el, er, menc, denom, h, out, E);
}